// DirectHorizontalLineFilter_71073118814671
// MI455X (gfx1250) — compile-verified
//
#include <hip/hip_runtime.h>
#include <hip/hip_bf16.h>

typedef __attribute__((ext_vector_type(2))) float v2f;
typedef __attribute__((ext_vector_type(8))) float v8f;

#define HH 512
#define WW 512
#define NPLANES 512            // B*C = 8*64
#define NROWS   (NPLANES * HH) // 262144

// ---------------- Kernel 1: per-row std -> soft line mask -------------------
// One wave32 per row. Lane l loads float4 at cols 4*(l+32*i), i=0..3 (fully
// coalesced B128), reduces sum/sumsq across the wave with shfl_xor.
__global__ void row_mask_kernel(const float* __restrict__ x,
                                float* __restrict__ mask) {
    const int wave = threadIdx.x >> 5;
    const int lane = threadIdx.x & 31;
    const long long row = (long long)blockIdx.x * 8 + wave;
    const float* xr = x + row * WW;

    float sum = 0.0f, sq = 0.0f;
#pragma unroll
    for (int i = 0; i < 4; ++i) {
        const float4 v = ((const float4*)xr)[lane + 32 * i];
        sum += v.x + v.y + v.z + v.w;
        sq  += v.x * v.x + v.y * v.y + v.z * v.z + v.w * v.w;
    }
#pragma unroll
    for (int off = 16; off >= 1; off >>= 1) {
        sum += __shfl_xor(sum, off, 32);
        sq  += __shfl_xor(sq,  off, 32);
    }
    if (lane == 0) {
        const float mean = sum * (1.0f / 512.0f);
        float var = (sq - sum * mean) * (1.0f / 511.0f);  // ddof=1
        var = var < 0.0f ? 0.0f : var;
        const float s = sqrtf(var);
        // sigmoid((0.05 - s)*10) = 1/(1+exp((s-0.05)*10))
        mask[row] = 1.0f / (1.0f + expf((s - 0.05f) * 10.0f));
    }
}

// ---------------- Kernel 2: 5-tap box of mask along H, * STRENGTH ----------
__global__ void smooth_mask_kernel(const float* __restrict__ mask,
                                   float* __restrict__ wgt) {
    const int idx = blockIdx.x * blockDim.x + threadIdx.x;  // 0..NROWS-1
    const int h    = idx & (HH - 1);
    const int base = idx - h;
    float s = 0.0f;
#pragma unroll
    for (int k = -2; k <= 2; ++k) {
        int hh = h + k;
        hh = hh < 0 ? 0 : (hh > HH - 1 ? HH - 1 : hh);
        s += mask[base + hh];
    }
    wgt[idx] = s * (0.8f * 0.2f);  // STRENGTH * mean
}

// ---------------- Kernel 3: vertical box via f32 WMMA + blend ---------------
// One wave per 16x16 output tile. VS[16,16] = sum_{g=0..4} A_g[16,4] * B_g[4,16]
// with A_g the 0.2-banded matrix chunk and B_g rows = clamped input rows
// (clamping == replicate padding). Layouts per CDNA5 ISA 7.12.2:
//   A 16x4 : lane m=L&15;  VGPR0=K(2*half), VGPR1=K(2*half+1)
//   B 4x16 : lane n=L&15;  VGPR0=row(2*half), VGPR1=row(2*half+1)
//   C/D    : VGPR v: M = v + 8*half, N = L&15
__global__ void blend_kernel(const float* __restrict__ x,
                             const float* __restrict__ wgt,
                             float* __restrict__ out) {
    const int wave = threadIdx.x >> 5;
    const int lane = threadIdx.x & 31;
    const int tile = blockIdx.x * 8 + wave;   // 0 .. 524287
    const int p  = tile >> 10;                // plane
    const int t  = tile & 1023;
    const int h0 = (t >> 5) << 4;
    const int w0 = (t & 31) << 4;

    const float* xp = x   + (long long)p * (HH * WW);
    const float* wp = wgt + p * HH;
    float*       op = out + (long long)p * (HH * WW);

    const int m    = lane & 15;   // A row / C-D column index bits
    const int half = lane >> 4;
    const int n    = m;           // B/C/D column for this lane

    v8f c = {};
#pragma unroll
    for (int g = 0; g < 5; ++g) {
        const int j0 = 4 * g + 2 * half;  // K indices {j0, j0+1} of the 20-band
        v2f a;
        a.x = (j0     >= m && j0     <= m + 4) ? 0.2f : 0.0f;
        a.y = (j0 + 1 >= m && j0 + 1 <= m + 4) ? 0.2f : 0.0f;

        int rA = h0 - 2 + j0;
        int rB = rA + 1;
        rA = rA < 0 ? 0 : (rA > HH - 1 ? HH - 1 : rA);
        rB = rB < 0 ? 0 : (rB > HH - 1 ? HH - 1 : rB);
        v2f b;
        b.x = xp[rA * WW + w0 + n];
        b.y = xp[rB * WW + w0 + n];

        // (neg_a, A, neg_b, B, c_mod, C, reuse_a, reuse_b)
        c = __builtin_amdgcn_wmma_f32_16x16x4_f32(false, a, false, b,
                                                  (short)0, c, false, false);
    }

    // Blend: out = x + (vs - x) * w
#pragma unroll
    for (int v = 0; v < 8; ++v) {
        const int row = h0 + v + 8 * half;
        const float xv = xp[row * WW + w0 + n];
        const float wv = wp[row];
        op[row * WW + w0 + n] = xv + (c[v] - xv) * wv;
    }
}

extern "C" void kernel_launch(void* const* d_in, const int* in_sizes, int n_in,
                              void* d_out, int out_size, void* d_ws, size_t ws_size,
                              hipStream_t stream) {
    const float* x = (const float*)d_in[0];
    float* out  = (float*)d_out;
    float* mask = (float*)d_ws;            // NROWS floats = 1 MiB
    float* wgt  = mask + NROWS;            // NROWS floats = 1 MiB

    // 1) per-row mask: 262144 rows, 8 waves (rows) per 256-thread block
    row_mask_kernel<<<NROWS / 8, 256, 0, stream>>>(x, mask);
    // 2) smooth mask along H
    smooth_mask_kernel<<<NROWS / 256, 256, 0, stream>>>(mask, wgt);
    // 3) WMMA vertical box + blend: 512*32*32 = 524288 tiles, 8 per block
    blend_kernel<<<(NPLANES * 32 * 32) / 8, 256, 0, stream>>>(x, wgt, out);
}